// ROIPooling_29600914604506
// MI455X (gfx1250) — compile-verified
//
#include <hip/hip_runtime.h>
#include <hip/hip_bf16.h>
#include <math.h>

// ROI max-pool, shapes fixed by the reference: B=4,H=50,W=50,C=256,N=256,K=7.
// Memory-bound gather+max: no MAC structure -> WMMA inapplicable. Optimized for
// MI455X memory system: b128 coalesced loads (input is L2-resident, 10.24MB of
// 192MB), 64-thread (2-wave) blocks, gfx1250 global_prefetch_b8 for next-row first
// touch, and non-temporal b128 stores for the write-once output.

typedef float v4f __attribute__((ext_vector_type(4)));

#define K_ 7
#define B_ 4
#define H_ 50
#define W_ 50
#define C_ 256
#define N_ 256
#define C4_ (C_ / 4)   // 64 float4 per pixel; one per lane of a 64-thread block

static __device__ __forceinline__ v4f vmax4(v4f a, v4f b) {
    v4f r;
    r.x = fmaxf(a.x, b.x);
    r.y = fmaxf(a.y, b.y);
    r.z = fmaxf(a.z, b.z);
    r.w = fmaxf(a.w, b.w);
    return r;
}

__global__ __launch_bounds__(64) void roi_maxpool_kernel(
    const float* __restrict__ x,
    const int*   __restrict__ roi_batch,
    const int*   __restrict__ roi_boxes,
    float*       __restrict__ out)
{
    const int tid = threadIdx.x;            // lane handles 4 consecutive channels
    const int bin = blockIdx.x;             // bin = (n*K + ky)*K + kx  (matches out layout)
    const int n   = bin / (K_ * K_);
    const int rem = bin - n * (K_ * K_);
    const int ky  = rem / K_;
    const int kx  = rem - ky * K_;

    const int b  = roi_batch[n];
    const int y1 = roi_boxes[4 * n + 0];
    const int x1 = roi_boxes[4 * n + 1];
    const int y2 = roi_boxes[4 * n + 2];
    const int x2 = roi_boxes[4 * n + 3];
    const int rh = (y2 - y1) > 1 ? (y2 - y1) : 1;
    const int rw = (x2 - x1) > 1 ? (x2 - x1) : 1;

    // bin ky covers relative rows [ceil(ky*rh/K), ceil((ky+1)*rh/K)), clamped to [y1,y2)
    int rs = y1 + ( ky      * rh + (K_ - 1)) / K_;
    int re = y1 + ((ky + 1) * rh + (K_ - 1)) / K_;
    if (re > y2) re = y2;
    int cs = x1 + ( kx      * rw + (K_ - 1)) / K_;
    int ce = x1 + ((kx + 1) * rw + (K_ - 1)) / K_;
    if (ce > x2) ce = x2;

    v4f acc0 = (v4f)(-INFINITY);
    v4f acc1 = acc0;

    if (rs < re && cs < ce) {
        const v4f* xv = (const v4f*)x;
        for (int row = rs; row < re; ++row) {
            const v4f* rowp = xv + (long)(b * H_ + row) * W_ * C4_;
            if (row + 1 < re) {
                // gfx1250 prefetch of next row's first bin column (whole 1KB
                // channel vector covered across the 64 lanes, 16B apart).
                const v4f* nextp = rowp + (long)W_ * C4_;
                __builtin_prefetch((const void*)&nextp[(long)cs * C4_ + tid], 0, 1);
            }
            int col = cs;
            for (; col + 1 < ce; col += 2) {           // 2-way unroll: 2 b128 loads in flight
                v4f v0 = rowp[(long)col       * C4_ + tid];
                v4f v1 = rowp[(long)(col + 1) * C4_ + tid];
                acc0 = vmax4(acc0, v0);
                acc1 = vmax4(acc1, v1);
            }
            if (col < ce) {
                v4f v0 = rowp[(long)col * C4_ + tid];
                acc0 = vmax4(acc0, v0);
            }
        }
    }

    v4f m = vmax4(acc0, acc1);
    // Reference washes non-finite (empty-bin -inf) to 0.
    v4f res;
    res.x = __builtin_isfinite(m.x) ? m.x : 0.0f;
    res.y = __builtin_isfinite(m.y) ? m.y : 0.0f;
    res.z = __builtin_isfinite(m.z) ? m.z : 0.0f;
    res.w = __builtin_isfinite(m.w) ? m.w : 0.0f;

    // Write-once streaming output: non-temporal b128 store, keep L2 for x.
    __builtin_nontemporal_store(res, (v4f*)out + (long)bin * C4_ + tid);
}

extern "C" void kernel_launch(void* const* d_in, const int* in_sizes, int n_in,
                              void* d_out, int out_size, void* d_ws, size_t ws_size,
                              hipStream_t stream) {
    (void)in_sizes; (void)n_in; (void)out_size; (void)d_ws; (void)ws_size;
    const float* x         = (const float*)d_in[0];   // (4,50,50,256) f32
    const int*   roi_batch = (const int*)d_in[1];     // (256,) i32
    const int*   roi_boxes = (const int*)d_in[2];     // (256,4) i32 [y1,x1,y2,x2]
    float*       out       = (float*)d_out;           // (256,7,7,256) f32

    const int nbins = N_ * K_ * K_;                   // 12544 blocks, 2 waves each
    roi_maxpool_kernel<<<dim3(nbins), dim3(64), 0, stream>>>(x, roi_batch, roi_boxes, out);
}